// QuaternionMaxAmpPool2d_7748121002663
// MI455X (gfx1250) — compile-verified
//
#include <hip/hip_runtime.h>
#include <cstdint>

// QuaternionMaxAmpPool2d: x (16,256,128,128) f32 -> out (16,256,64,64) f32
// Pure streaming op: 268 MB read + 67 MB write => ~14.4 us at 23.3 TB/s HBM.
// CDNA5 path: double-buffered GLOBAL_LOAD_ASYNC_TO_LDS_B128 pipeline with
// S_WAIT_ASYNCCNT (ASYNCcnt), per CDNA5 ISA. All traffic
// is touch-once (335 MB vs 192 MB L2), so loads/stores use non-temporal
// hints (CDNA5 ISA) to avoid cache pollution.
// WMMA is not applicable (argmax/select, no contraction); sqrt dropped
// (monotone under argmax).

#define TPB   256
#define ITERS 4

constexpr int CB  = 256;   // channels
constexpr int C4  = 64;    // channels per quaternion component
constexpr int Hh  = 128;
constexpr int Ww  = 128;
constexpr int H2  = 64;
constexpr int W2  = 64;
constexpr int OW2 = 32;    // output-column PAIRS per row

typedef float v2f __attribute__((ext_vector_type(2)));

__device__ __forceinline__ uint32_t lds_offset(const void* p) {
  // Generic LDS pointer = {shared-aperture in high dword, LDS byte offset in
  // low dword}; DS / async-LDS instructions take the low-32 offset.
  return (uint32_t)(uintptr_t)p;
}

__device__ __forceinline__ void async_ld_b128_nt(uint32_t lds, const float* g) {
  asm volatile("global_load_async_to_lds_b128 %0, %1, off th:TH_LOAD_NT"
               :: "v"(lds), "v"((uint64_t)(uintptr_t)g)
               : "memory");
}

__global__ __launch_bounds__(TPB) void qpool_kernel(const float* __restrict__ x,
                                                    float* __restrict__ out) {
  __shared__ float4 smem[2][TPB * 8];   // 2 x 32 KB double buffer
  const int tid = threadIdx.x;
  const int u0  = blockIdx.x * (TPB * ITERS) + tid;

  uint32_t lbase[2];
  lbase[0] = lds_offset(&smem[0][tid * 8]);
  lbase[1] = lds_offset(&smem[1][tid * 8]);

  auto issue = [&](int i, int buf) {
    const int u   = u0 + i * TPB;
    const int ow2 = u & (OW2 - 1);
    const int oh  = (u >> 5) & (H2 - 1);
    const int cc  = (u >> 11) & (C4 - 1);
    const int b   = u >> 17;
    const float* base = x + (size_t)(b * CB + cc) * (Hh * Ww)
                          + (2 * oh) * Ww + 4 * ow2;
    uint32_t l = lbase[buf];
#pragma unroll
    for (int q = 0; q < 4; ++q) {
      const float* cb = base + (size_t)q * (C4 * Hh * Ww);
      async_ld_b128_nt(l,      cb);       // row 2*oh   : 4 floats
      async_ld_b128_nt(l + 16, cb + Ww);  // row 2*oh+1 : 4 floats
      l += 32;
    }
  };

  issue(0, 0);

  for (int i = 0; i < ITERS; ++i) {
    const int buf = i & 1;
    if (i + 1 < ITERS) {
      issue(i + 1, buf ^ 1);
      // 8 loads of the next batch outstanding; <=8 means current batch landed
      asm volatile("s_wait_asynccnt 0x8" ::: "memory");
    } else {
      asm volatile("s_wait_asynccnt 0x0" ::: "memory");
    }

    const float4* s = &smem[buf][tid * 8];
    float4 r0[4], r1[4];
#pragma unroll
    for (int q = 0; q < 4; ++q) { r0[q] = s[2 * q]; r1[q] = s[2 * q + 1]; }

    const int u   = u0 + i * TPB;
    const int ow2 = u & (OW2 - 1);
    const int oh  = (u >> 5) & (H2 - 1);
    const int cc  = (u >> 11) & (C4 - 1);
    const int b   = u >> 17;

    float res[2][4];
#pragma unroll
    for (int j = 0; j < 2; ++j) {
      // window positions p: (r0,c0),(r0,c1),(r1,c0),(r1,c1) == kh*2+kw order
      float c0[4], c1[4], c2[4], c3[4];
      float a0 = 0.f, a1 = 0.f, a2 = 0.f, a3 = 0.f;
#pragma unroll
      for (int q = 0; q < 4; ++q) {
        const float p0 = j ? r0[q].z : r0[q].x;
        const float p1 = j ? r0[q].w : r0[q].y;
        const float p2 = j ? r1[q].z : r1[q].x;
        const float p3 = j ? r1[q].w : r1[q].y;
        c0[q] = p0; c1[q] = p1; c2[q] = p2; c3[q] = p3;
        a0 = fmaf(p0, p0, a0);
        a1 = fmaf(p1, p1, a1);
        a2 = fmaf(p2, p2, a2);
        a3 = fmaf(p3, p3, a3);
      }
      // branchless first-occurrence argmax (strict >), matches jnp.argmax
      float best = a0;
      float o[4];
#pragma unroll
      for (int q = 0; q < 4; ++q) o[q] = c0[q];
      const bool t1 = a1 > best; best = t1 ? a1 : best;
#pragma unroll
      for (int q = 0; q < 4; ++q) o[q] = t1 ? c1[q] : o[q];
      const bool t2 = a2 > best; best = t2 ? a2 : best;
#pragma unroll
      for (int q = 0; q < 4; ++q) o[q] = t2 ? c2[q] : o[q];
      const bool t3 = a3 > best;
#pragma unroll
      for (int q = 0; q < 4; ++q) o[q] = t3 ? c3[q] : o[q];
#pragma unroll
      for (int q = 0; q < 4; ++q) res[j][q] = o[q];
    }

    float* obase = out + (size_t)(b * CB + cc) * (H2 * W2) + oh * W2 + 2 * ow2;
#pragma unroll
    for (int q = 0; q < 4; ++q) {
      v2f v; v.x = res[0][q]; v.y = res[1][q];
      // touch-once output: non-temporal store (th:TH_STORE_NT), coalesced b64
      __builtin_nontemporal_store(v, (v2f*)(obase + (size_t)q * (C4 * H2 * W2)));
    }
  }
}

extern "C" void kernel_launch(void* const* d_in, const int* in_sizes, int n_in,
                              void* d_out, int out_size, void* d_ws, size_t ws_size,
                              hipStream_t stream) {
  const float* x = (const float*)d_in[0];
  float* out = (float*)d_out;
  constexpr int units  = 16 * C4 * H2 * OW2;      // 2,097,152 thread-units
  constexpr int blocks = units / (TPB * ITERS);   // 2048 blocks, exact grid
  qpool_kernel<<<blocks, TPB, 0, stream>>>(x, out);
}